// NeRFRenderer_20478404067764
// MI455X (gfx1250) — compile-verified
//
#include <hip/hip_runtime.h>
#include <cstdint>

#define T0   64
#define TF   128
#define RPB  64                 // rays per block == threads per block (2 waves)
#define NWAVE (RPB / 32)
#define OUTC (4 + 3 * TF)       // 388 floats per output row

typedef float v2f __attribute__((ext_vector_type(2)));
typedef float v8f __attribute__((ext_vector_type(8)));

__global__ __launch_bounds__(RPB) void nerf_render_kernel(
    const float* __restrict__ rays_o,
    const float* __restrict__ rays_d,
    const float* __restrict__ aabb,
    const float* __restrict__ wts,
    const float* __restrict__ sig,
    const float* __restrict__ rgb,
    float* __restrict__ out)
{
    // Staged coarse weights: rows padded to 68 (=> 272B, 16B aligned rows for b128 async DMA)
    __shared__ __align__(16) float s_wt[RPB][68];
    // Fine-sample weights (WMMA A operand): stride 130 floats (520B, 8B aligned, bank-conflict free)
    __shared__ __align__(16) float s_w[RPB][130];

    const int t    = threadIdx.x;
    const int lane = t & 31;
    const int wave = t >> 5;
    const int B0   = blockIdx.x * RPB;
    const int ray  = B0 + t;

    // ---------------- Phase 0: async DMA of weights_coarse -> LDS ----------------
    {
        const unsigned ldsBase = (unsigned)(uintptr_t)(&s_wt[0][0]);
        const float*   gbase   = wts + (size_t)B0 * T0;
        #pragma unroll
        for (int it = 0; it < T0 / 4; ++it) {          // 16 x (64 lanes * 16B) = 16KB
            int      p    = it * RPB + t;              // 16-byte packet id
            int      flat = p * 4;                     // flat float index (rows contiguous)
            int      r    = flat >> 6;                 // ray-in-block
            int      c    = flat & 63;                 // column (multiple of 4)
            unsigned lds  = ldsBase + (unsigned)((r * 68 + c) * 4);
            const float* g = gbase + flat;
            asm volatile("global_load_async_to_lds_b128 %0, %1, off"
                         :: "v"(lds), "v"(g) : "memory");
        }
        asm volatile("s_wait_asynccnt 0x0" ::: "memory");
    }
    __syncthreads();

    // ---------------- Phase A: per-ray sequential pipeline (thread = ray) --------
    const float ox = rays_o[ray * 3 + 0], oy = rays_o[ray * 3 + 1], oz = rays_o[ray * 3 + 2];
    const float dx = rays_d[ray * 3 + 0], dy = rays_d[ray * 3 + 1], dz = rays_d[ray * 3 + 2];

    float nearv, farv;
    {
        float t0x = (aabb[0] - ox) / (dx + 1e-15f), t1x = (aabb[3] - ox) / (dx + 1e-15f);
        float t0y = (aabb[1] - oy) / (dy + 1e-15f), t1y = (aabb[4] - oy) / (dy + 1e-15f);
        float t0z = (aabb[2] - oz) / (dz + 1e-15f), t1z = (aabb[5] - oz) / (dz + 1e-15f);
        nearv = fmaxf(fmaxf(fminf(t0x, t1x), fminf(t0y, t1y)), fminf(t0z, t1z));
        farv  = fminf(fminf(fmaxf(t0x, t1x), fmaxf(t0y, t1y)), fmaxf(t0z, t1z));
        if (farv < nearv) { nearv = 1e9f; farv = 1e9f; }
        nearv = fmaxf(nearv, 0.05f);
    }
    const float binsize = (farv - nearv) * (1.0f / (float)T0);

    // sum of (w + 0.01)
    float sumW = 0.f;
    #pragma unroll 8
    for (int i = 0; i < T0; ++i) sumW += s_wt[t][i] + 0.01f;
    const float invSum = 1.0f / sumW;

    const float* sigRow = sig + (size_t)ray * TF;
    float*       outRow = out + (size_t)ray * OUTC;

    // two-pointer merge over monotone cdf / u; stream edges -> samples
    int   ib    = 0;
    float run   = s_wt[t][0] + 0.01f;
    float cLo   = 0.f;
    float cHi   = fminf(run * invSum, 1.f);
    float trans = 1.f;
    float depth = 0.f;
    float prevE = 0.f;

    for (int j = 0; j <= TF; ++j) {
        const float u = ((float)j + 0.5f) * (1.0f / 129.0f);   // linspace(.5/129, 1-.5/129, 129)
        while (cHi <= u && ib < T0) {
            ++ib;
            cLo = cHi;
            if (ib < T0) { run += s_wt[t][ib] + 0.01f; cHi = fminf(run * invSum, 1.f); }
            else         { cHi = 3.0e38f; }
        }
        const int   above = (ib + 1 < T0 + 1) ? (ib + 1) : T0;
        const float b0    = nearv + binsize * (float)ib;
        const float b1    = nearv + binsize * (float)above;
        const float den   = cHi - cLo;
        float tt = 0.f;
        if (den > 0.f) tt = fminf(fmaxf((u - cLo) / fmaxf(den, 1e-12f), 0.f), 1.f);
        const float edge = b0 + tt * (b1 - b0);

        if (j > 0) {
            const int   k     = j - 1;
            const float delta = edge - prevE;
            const float zm    = 0.5f * (edge + prevE);

            __builtin_prefetch(sigRow + ((k + 8) & (TF - 1)), 0, 0);  // global_prefetch_b8
            const float sgm   = sigRow[k];
            const float alpha = 1.f - __expf(-sgm * delta);
            const float wk    = alpha * trans;
            trans             = trans * (1.f - alpha + 1e-10f);

            s_w[t][k] = wk;            // WMMA A operand
            depth     = fmaf(wk, zm, depth);

            // sample position + mip-NeRF-360 contraction
            const float px = fmaf(dx, zm, ox), py = fmaf(dy, zm, oy), pz = fmaf(dz, zm, oz);
            const float ax = fabsf(px), ay = fabsf(py), az = fabsf(pz);
            const float mag = fmaxf(ax, fmaxf(ay, az));
            float cx = px, cy = py, cz = pz;
            if (mag >= 1.f) {
                const float inv = 1.f / mag;
                const float sM  = (2.f - inv) * inv;
                const int   mx  = (ax >= ay) ? ((ax >= az) ? 0 : 2) : ((ay >= az) ? 1 : 2);
                cx = px * ((mx == 0) ? sM : inv);
                cy = py * ((mx == 1) ? sM : inv);
                cz = pz * ((mx == 2) ? sM : inv);
            }
            outRow[4 + 3 * k + 0] = cx;
            outRow[4 + 3 * k + 1] = cy;
            outRow[4 + 3 * k + 2] = cz;
        }
        prevE = edge;
    }
    outRow[3] = depth;
    __syncthreads();

    // ---------------- Phase B: image = diag(W x RGB^T) via FP32 WMMA -------------
    // 4 tiles of 16 rays x 3 channels = 12 chains; each chain: 32x V_WMMA_F32_16X16X4_F32.
    // A[i][k] = w[tile ray i][k] (LDS), B[k][j] = rgb[tile ray j][k][ch] (global, each
    // element touched exactly once). Desired outputs are the diagonal of D.
    for (int chain = wave; chain < (RPB / 16) * 3; chain += NWAVE) {
        const int tile = chain / 3;
        const int ch   = chain % 3;
        const int tb   = tile * 16;
        const int rI   = tb + (lane & 15);
        const int kOff = (lane >> 4) << 1;          // lanes 0-15 -> K+0/1, lanes 16-31 -> K+2/3
        const float* bptr = rgb + ((size_t)(B0 + rI) * TF) * 3 + ch;

        v8f acc = {0.f, 0.f, 0.f, 0.f, 0.f, 0.f, 0.f, 0.f};
        #pragma unroll
        for (int k0 = 0; k0 < TF; k0 += 4) {
            const int ka = k0 + kOff;
            v2f a = *(const v2f*)(&s_w[rI][ka]);    // 8B-aligned ds_load_b64
            v2f b;
            b.x = bptr[ka * 3];
            b.y = bptr[ka * 3 + 3];
            acc = __builtin_amdgcn_wmma_f32_16x16x4_f32(
                false, a, false, b, (short)0, acc, false, false);
        }
        // diagonal extraction: row M<8 lives in VGPR M lanes 0-15 (N=lane);
        // row M>=8 in VGPR M-8 lanes 16-31 (N=lane-16) -> (i,i) at lane i / i+16.
        #pragma unroll
        for (int i = 0; i < 16; ++i) {
            const int src = (i < 8) ? i : (i + 16);
            if (lane == src)
                out[(size_t)(B0 + tb + i) * OUTC + ch] = acc[i & 7];
        }
    }
}

extern "C" void kernel_launch(void* const* d_in, const int* in_sizes, int n_in,
                              void* d_out, int out_size, void* d_ws, size_t ws_size,
                              hipStream_t stream) {
    const float* rays_o = (const float*)d_in[0];
    const float* rays_d = (const float*)d_in[1];
    const float* aabb   = (const float*)d_in[2];
    const float* wts    = (const float*)d_in[3];
    const float* sig    = (const float*)d_in[4];
    const float* rgb    = (const float*)d_in[5];
    float* out = (float*)d_out;

    const int N      = in_sizes[0] / 3;   // 65536
    const int blocks = N / RPB;
    nerf_render_kernel<<<blocks, RPB, 0, stream>>>(rays_o, rays_d, aabb, wts, sig, rgb, out);
}